// MoGPrior_46823733461752
// MI455X (gfx1250) — compile-verified
//
#include <hip/hip_runtime.h>
#include <cstdint>

// Problem constants (fixed by reference setup_inputs): MC=16, B=4096, K=256, D=128
#define MC_N   16
#define B_N    4096
#define K_N    256
#define D_N    128
#define KC     32                 // K-chunk staged in LDS: 32*128*3 floats = 48 KB
#define CHUNK_FLTS (3 * KC * D_N) // floats per staged chunk (one buffer)
#define LOG2PI_F 1.8378770664093453f
#define L2E_F    1.4426950408889634f   // log2(e)
#define LN2_F    0.6931471805599453f

#if __has_builtin(__builtin_amdgcn_tensor_load_to_lds)
#define HAVE_TDM 1
#endif

typedef unsigned int u32x4 __attribute__((ext_vector_type(4)));
typedef int          i32x4 __attribute__((ext_vector_type(4)));
typedef int          i32x8 __attribute__((ext_vector_type(8)));

// ---------------------------------------------------------------------------
// TDM helper: stage one K-chunk (3 rows: c0,c1,c2 spaced K_N*D_N floats apart,
// each row KC*D_N contiguous floats) global -> LDS at byte offset lds_off.
// Caller gates to a single lane (TDM ignores EXEC; other waves branch around).
// ---------------------------------------------------------------------------
__device__ __forceinline__ void tdm_stage(const float* __restrict__ gsrc,
                                          unsigned lds_off) {
#if defined(HAVE_TDM)
    const unsigned long long ga = (unsigned long long)(uintptr_t)gsrc;
    u32x4 g0;
    g0[0] = 1u;                                        // count=1, user mode
    g0[1] = (unsigned)__builtin_amdgcn_groupstaticsize() + lds_off;
    g0[2] = (unsigned)(ga & 0xFFFFFFFFu);              // global_addr lo
    g0[3] = (unsigned)(ga >> 32) | (2u << 30);         // global_addr hi | type=2
    i32x8 g1;
    g1[0] = (int)(2u << 16);                           // data_size=2 (4 bytes)
    g1[1] = (int)((unsigned)(KC * D_N) << 16);         // tensor_dim0 lo16
    g1[2] = (int)(3u << 16);                           // dim0 hi=0 | tensor_dim1=3
    g1[3] = (int)((unsigned)(KC * D_N) << 16);         // dim1 hi=0 | tile_dim0
    g1[4] = 3;                                         // tile_dim1=3, tile_dim2=0
    g1[5] = (int)(K_N * D_N);                          // tensor_dim0_stride=32768
    g1[6] = 0;
    g1[7] = 0;
    i32x4 g2 = {0, 0, 0, 0};
    i32x4 g3 = {0, 0, 0, 0};
#if defined(__clang_major__) && (__clang_major__ >= 23)
    i32x8 gz = {0, 0, 0, 0, 0, 0, 0, 0};
    __builtin_amdgcn_tensor_load_to_lds(g0, g1, g2, g3, gz, 0);
#else
    __builtin_amdgcn_tensor_load_to_lds(g0, g1, g2, g3, 0);
#endif
#else
    (void)gsrc; (void)lds_off;
#endif
}

__device__ __forceinline__ void tdm_wait() {
#if defined(HAVE_TDM)
    if (threadIdx.x == 0) {
#if __has_builtin(__builtin_amdgcn_s_wait_tensorcnt)
        __builtin_amdgcn_s_wait_tensorcnt(0);
#else
        asm volatile("s_wait_tensorcnt 0" ::: "memory");
#endif
    }
#endif
    __syncthreads();
}

#if !defined(HAVE_TDM)
// Cooperative fallback staging (only used if TDM builtin is unavailable)
__device__ __forceinline__ void stage_fallback(float* __restrict__ dst,
                                               const float* __restrict__ c0g,
                                               const float* __restrict__ c1g,
                                               const float* __restrict__ c2g,
                                               int kc) {
    for (int i = threadIdx.x; i < KC * D_N; i += 256) {
        const int gi = kc * D_N + i;
        dst[i]               = c0g[gi];
        dst[KC * D_N + i]    = c1g[gi];
        dst[2 * KC * D_N + i] = c2g[gi];
    }
}
#endif

// ---------------------------------------------------------------------------
// Main kernel: fused Ez/Ez2 + two-pass logsumexp over K.
// 256 threads = 8 wave32s. Thread owns a d-quad (4 consecutive d) x 2 rows.
// Block covers 16 b-rows x all 128 d. Coefficients double-buffered in LDS via
// the Tensor Data Mover, overlapped with compute.
// ---------------------------------------------------------------------------
__global__ __launch_bounds__(256) void k_main(const float* __restrict__ z,
                                              const float* __restrict__ c0g,
                                              const float* __restrict__ c1g,
                                              const float* __restrict__ c2g,
                                              float* __restrict__ out) {
    extern __shared__ float smem[];             // 2 * CHUNK_FLTS floats (96 KB)
    const int tq = threadIdx.x & 31;            // d-quad id -> d = tq*4
    const int rl = threadIdx.x >> 5;            // row lane 0..7
    const int b0 = blockIdx.x * 16;             // this block: rows b0 .. b0+15

    // Kick off chunk-0 DMA immediately; it overlaps the z streaming below.
#if defined(HAVE_TDM)
    if (threadIdx.x == 0) tdm_stage(c0g, 0u);
#else
    stage_fallback(smem, c0g, c1g, c2g, 0);
#endif

    // ---- Phase 1: Ez / Ez2 (b128 coalesced: one wave = 512B per (row,mc)) --
    float Ez[8], Ez2[8], m[8], s[8];
#pragma unroll
    for (int r = 0; r < 2; ++r) {
        const int b = b0 + r * 8 + rl;
        float e0 = 0.f, e1 = 0.f, e2 = 0.f, e3 = 0.f;
        float q0 = 0.f, q1 = 0.f, q2 = 0.f, q3 = 0.f;
        const float* zp = z + (size_t)b * D_N + tq * 4;
#pragma unroll
        for (int mc = 0; mc < MC_N; ++mc) {
            const float4 v = *(const float4*)(zp + (size_t)mc * (size_t)(B_N * D_N));
            e0 += v.x; q0 = fmaf(v.x, v.x, q0);
            e1 += v.y; q1 = fmaf(v.y, v.y, q1);
            e2 += v.z; q2 = fmaf(v.z, v.z, q2);
            e3 += v.w; q3 = fmaf(v.w, v.w, q3);
        }
        const float inv = 1.0f / MC_N;
        Ez[r * 4 + 0] = e0 * inv;  Ez2[r * 4 + 0] = q0 * inv;
        Ez[r * 4 + 1] = e1 * inv;  Ez2[r * 4 + 1] = q1 * inv;
        Ez[r * 4 + 2] = e2 * inv;  Ez2[r * 4 + 2] = q2 * inv;
        Ez[r * 4 + 3] = e3 * inv;  Ez2[r * 4 + 3] = q3 * inv;
    }
#pragma unroll
    for (int p = 0; p < 8; ++p) { m[p] = -3.4e38f; s[p] = 0.f; }

    tdm_wait();   // chunk 0 resident

    // ---- Phase 2: 16 chunk-iterations = 2 passes x 8 K-chunks --------------
    const int NCH = 16;
    for (int it = 0; it < NCH; ++it) {
        const int buf = it & 1;
        // Prefetch next chunk into the other buffer (overlaps compute).
        if (it + 1 < NCH) {
            const int kcn = ((it + 1) & 7) * KC;
#if defined(HAVE_TDM)
            if (threadIdx.x == 0)
                tdm_stage(c0g + (size_t)kcn * D_N,
                          (unsigned)(((it + 1) & 1) * CHUNK_FLTS * 4));
#else
            stage_fallback(smem + ((it + 1) & 1) * CHUNK_FLTS, c0g, c1g, c2g, kcn);
#endif
        }

        const float4* S0 = (const float4*)(smem + buf * CHUNK_FLTS);
        const float4* S1 = S0 + KC * (D_N / 4);
        const float4* S2 = S0 + 2 * KC * (D_N / 4);

        if (it < 8) {
            // pass 0: running max (FMA + max only, no exp)
#pragma unroll 4
            for (int k = 0; k < KC; ++k) {
                const float4 A0 = S0[k * 32 + tq];
                const float4 A1 = S1[k * 32 + tq];
                const float4 A2 = S2[k * 32 + tq];
                const float a0[4] = {A0.x, A0.y, A0.z, A0.w};
                const float a1[4] = {A1.x, A1.y, A1.z, A1.w};
                const float a2[4] = {A2.x, A2.y, A2.z, A2.w};
#pragma unroll
                for (int p = 0; p < 8; ++p) {
                    const int c = p & 3;
                    const float t = fmaf(a2[c], Ez2[p], fmaf(a1[c], Ez[p], a0[c]));
                    m[p] = fmaxf(m[p], t);
                }
            }
        } else {
            // pass 1: sum of exp2 (one v_exp_f32 per term, co-executes with FMA)
#pragma unroll 4
            for (int k = 0; k < KC; ++k) {
                const float4 A0 = S0[k * 32 + tq];
                const float4 A1 = S1[k * 32 + tq];
                const float4 A2 = S2[k * 32 + tq];
                const float a0[4] = {A0.x, A0.y, A0.z, A0.w};
                const float a1[4] = {A1.x, A1.y, A1.z, A1.w};
                const float a2[4] = {A2.x, A2.y, A2.z, A2.w};
#pragma unroll
                for (int p = 0; p < 8; ++p) {
                    const int c = p & 3;
                    const float t = fmaf(a2[c], Ez2[p], fmaf(a1[c], Ez[p], a0[c]));
                    s[p] += __builtin_amdgcn_exp2f(t - m[p]);
                }
            }
        }
        tdm_wait();   // next chunk resident; everyone done with this buffer
    }

    // ---- Epilogue: out = ln2 * (m + log2(s)), b128 stores ------------------
#pragma unroll
    for (int r = 0; r < 2; ++r) {
        const int b = b0 + r * 8 + rl;
        float4 o;
        o.x = LN2_F * (m[r * 4 + 0] + __builtin_amdgcn_logf(s[r * 4 + 0]));
        o.y = LN2_F * (m[r * 4 + 1] + __builtin_amdgcn_logf(s[r * 4 + 1]));
        o.z = LN2_F * (m[r * 4 + 2] + __builtin_amdgcn_logf(s[r * 4 + 2]));
        o.w = LN2_F * (m[r * 4 + 3] + __builtin_amdgcn_logf(s[r * 4 + 3]));
        *(float4*)(out + (size_t)b * D_N + tq * 4) = o;
    }
}

// ---------------------------------------------------------------------------
// Kernel: log_softmax of w (K=256 values) -> lw
// ---------------------------------------------------------------------------
__global__ __launch_bounds__(K_N) void k_logw(const float* __restrict__ w,
                                              float* __restrict__ lw) {
    __shared__ float red[K_N];
    const int t = threadIdx.x;
    const float v = w[t];
    red[t] = v;
    __syncthreads();
    for (int off = K_N / 2; off > 0; off >>= 1) {
        if (t < off) red[t] = fmaxf(red[t], red[t + off]);
        __syncthreads();
    }
    const float mx = red[0];
    __syncthreads();
    red[t] = __expf(v - mx);
    __syncthreads();
    for (int off = K_N / 2; off > 0; off >>= 1) {
        if (t < off) red[t] += red[t + off];
        __syncthreads();
    }
    const float lse = mx + __logf(red[0]);
    lw[t] = v - lse;
}

// ---------------------------------------------------------------------------
// Kernel: per-(k,d) coefficients, pre-scaled into log2 domain.
//   a2[k,b,d] = C0[k,d] + C1[k,d]*Ez + C2[k,d]*Ez2   (all * log2(e))
// ---------------------------------------------------------------------------
__global__ __launch_bounds__(256) void k_coef(const float* __restrict__ means,
                                              const float* __restrict__ lvars,
                                              const float* __restrict__ lw,
                                              float* __restrict__ c0,
                                              float* __restrict__ c1,
                                              float* __restrict__ c2) {
    const int i = blockIdx.x * blockDim.x + threadIdx.x;   // 0 .. K*D-1
    const int k = i >> 7;                                  // i / D_N
    const float mu = means[i];
    const float lv = lvars[i];
    const float p  = __expf(-lv);                          // exp(-log_var)
    const float C2 = -0.5f * p;
    const float C1 = p * mu;
    const float C0 = -0.5f * (LOG2PI_F + lv) - 0.5f * p * mu * mu + lw[k];
    c0[i] = C0 * L2E_F;
    c1[i] = C1 * L2E_F;
    c2[i] = C2 * L2E_F;
}

// ---------------------------------------------------------------------------
extern "C" void kernel_launch(void* const* d_in, const int* in_sizes, int n_in,
                              void* d_out, int out_size, void* d_ws, size_t ws_size,
                              hipStream_t stream) {
    (void)in_sizes; (void)n_in; (void)out_size; (void)ws_size;
    const float* z     = (const float*)d_in[0];   // (16, 4096, 128)
    const float* means = (const float*)d_in[1];   // (256, 128)
    const float* lvars = (const float*)d_in[2];   // (256, 128)
    const float* w     = (const float*)d_in[3];   // (256,)
    float* out = (float*)d_out;                   // (4096, 128)

    float* ws = (float*)d_ws;
    float* lw = ws;                 // 256 floats
    float* c0 = ws + 256;           // 32768 floats
    float* c1 = c0 + K_N * D_N;     // 32768 floats  (contiguity required: TDM
    float* c2 = c1 + K_N * D_N;     //  row stride between c0/c1/c2 = 32768 elems)

    k_logw<<<1, K_N, 0, stream>>>(w, lw);
    k_coef<<<(K_N * D_N) / 256, 256, 0, stream>>>(means, lvars, lw, c0, c1, c2);
    k_main<<<B_N / 16, 256, 2 * CHUNK_FLTS * sizeof(float), stream>>>(z, c0, c1, c2, out);
}